// DenseGINConv_20804821582053
// MI455X (gfx1250) — compile-verified
//
#include <hip/hip_runtime.h>

#define D 128
#define ROWS 64          // rows (M) per block
#define KCH 32           // K rows of W staged in LDS per chunk

typedef __attribute__((ext_vector_type(2))) float v2f;
typedef __attribute__((ext_vector_type(4))) float f4v;
typedef __attribute__((ext_vector_type(8))) float v8f;
typedef int i4v __attribute__((vector_size(16)));   // matches builtin param type

#define AS1 __attribute__((address_space(1)))
#define AS3 __attribute__((address_space(3)))

#if defined(__has_builtin)
#  if __has_builtin(__builtin_amdgcn_global_load_async_to_lds_b128) && \
      __has_builtin(__builtin_amdgcn_s_wait_asynccnt)
#    define HAVE_ASYNC_LDS 1
#  endif
#endif

// ---------------------------------------------------------------------------
// Kernel 1: out[i,d] = (1 + eps) * x[i,d]          (d_out used as h staging)
// ---------------------------------------------------------------------------
__global__ __launch_bounds__(256) void gin_init(const float* __restrict__ x,
                                                const float* __restrict__ eps,
                                                float* __restrict__ out,
                                                int total4) {
  int idx = blockIdx.x * 256 + threadIdx.x;
  if (idx >= total4) return;
  float s = 1.0f + eps[0];
  f4v v = ((AS1 const f4v*)x)[idx];
  v *= s;
  ((AS1 f4v*)out)[idx] = v;
}

// ---------------------------------------------------------------------------
// Kernel 2: scatter-add  out[row] += val * x[col]   (one wave32 per edge)
// x rows (51 MB) stay L2-resident; atomics resolve in L2.
// ---------------------------------------------------------------------------
__global__ __launch_bounds__(256) void gin_scatter(const int* __restrict__ erow,
                                                   const int* __restrict__ ecol,
                                                   const float* __restrict__ eval,
                                                   const float* __restrict__ x,
                                                   float* out, int nedges) {
  int e    = (blockIdx.x * 256 + threadIdx.x) >> 5;
  int lane = threadIdx.x & 31;
  if (e >= nedges) return;
  int   r = erow[e];
  int   c = ecol[e];
  float w = eval[e];
  f4v v = ((AS1 const f4v*)(x + (size_t)c * D))[lane];   // 512B per row
  float* o = out + (size_t)r * D + lane * 4;
  atomicAdd(o + 0, w * v.x);
  atomicAdd(o + 1, w * v.y);
  atomicAdd(o + 2, w * v.z);
  atomicAdd(o + 3, w * v.w);
}

// ---------------------------------------------------------------------------
// Kernel 3: fused  relu(relu( relu(h@W1+b1) @ W2 + b2 ) + bias)
// 64 rows/block, 8 waves; wave w owns output cols [16w,16w+16) over four
// 16-row M-tiles (one B operand feeds 4 WMMAs). W staged in 32x128 LDS
// chunks via async-to-LDS; h tile staged once and rewritten in place with
// layer-1 output.
// ---------------------------------------------------------------------------
__global__ __launch_bounds__(256) void gin_mlp(float* hio,   // aliases d_out
                                               const float* __restrict__ W1,
                                               const float* __restrict__ b1,
                                               const float* __restrict__ W2,
                                               const float* __restrict__ b2,
                                               const float* __restrict__ bias,
                                               int n) {
  __shared__ float sH[ROWS][D + 4];   // stride 132: conflict-free, 16B-aligned
  __shared__ float sW[KCH][D];

  const int tid  = threadIdx.x;
  const int wave = tid >> 5;
  const int lane = tid & 31;
  const int ln   = lane & 15;
  const int hi   = lane >> 4;
  const int n0   = wave * 16;
  const int row0 = blockIdx.x * ROWS;

  // ---- stage input h tile: 64*128/4 = 2048 xb128, 8 per thread ----
#if HAVE_ASYNC_LDS
  #pragma unroll
  for (int i = 0; i < 8; ++i) {
    int idx = tid + i * 256;
    int r   = idx >> 5;
    int c4  = (idx & 31) * 4;
    int g   = row0 + r; if (g > n - 1) g = n - 1;
    __builtin_amdgcn_global_load_async_to_lds_b128(
        (AS1 i4v*)(hio + (size_t)g * D + c4),
        (AS3 i4v*)&sH[r][c4], 0, 0);
  }
  __builtin_amdgcn_s_wait_asynccnt(0);
#else
  {
    f4v t[8];
    #pragma unroll
    for (int i = 0; i < 8; ++i) {
      int idx = tid + i * 256;
      int g   = row0 + (idx >> 5); if (g > n - 1) g = n - 1;
      t[i] = *(AS1 const f4v*)(hio + (size_t)g * D + (idx & 31) * 4);
    }
    #pragma unroll
    for (int i = 0; i < 8; ++i) {
      int idx = tid + i * 256;
      *(f4v*)&sH[idx >> 5][(idx & 31) * 4] = t[i];
    }
  }
#endif

  v8f acc[4];
  const v8f vzero = {};

  for (int layer = 0; layer < 2; ++layer) {
    const float* Wp = (layer == 0) ? W1 : W2;
    #pragma unroll
    for (int t = 0; t < 4; ++t) acc[t] = vzero;

    for (int ks = 0; ks < D; ks += KCH) {
      __syncthreads();                          // sW free (and sH staged)
      // ---- stage 32x128 weight chunk: 1024 xb128, 4 per thread ----
#if HAVE_ASYNC_LDS
      #pragma unroll
      for (int i = 0; i < 4; ++i) {
        int idx = tid + i * 256;
        int kr  = idx >> 5;
        int c4  = (idx & 31) * 4;
        __builtin_amdgcn_global_load_async_to_lds_b128(
            (AS1 i4v*)(Wp + (size_t)(ks + kr) * D + c4),
            (AS3 i4v*)&sW[kr][c4], 0, 0);
      }
      __builtin_amdgcn_s_wait_asynccnt(0);
#else
      {
        f4v t[4];
        #pragma unroll
        for (int i = 0; i < 4; ++i) {
          int idx = tid + i * 256;
          t[i] = *(AS1 const f4v*)(Wp + (size_t)(ks + (idx >> 5)) * D +
                                   (idx & 31) * 4);
        }
        #pragma unroll
        for (int i = 0; i < 4; ++i) {
          int idx = tid + i * 256;
          *(f4v*)&sW[idx >> 5][(idx & 31) * 4] = t[i];
        }
      }
#endif
      __syncthreads();

      #pragma unroll
      for (int kk = 0; kk < KCH; kk += 4) {
        int kb = kk + hi * 2;                   // this lane's K pair
        v2f bm;
        bm.x = sW[kb][n0 + ln];
        bm.y = sW[kb + 1][n0 + ln];
        #pragma unroll
        for (int mt = 0; mt < 4; ++mt) {
          v2f am = *(const v2f*)&sH[mt * 16 + ln][ks + kb];
          acc[mt] = __builtin_amdgcn_wmma_f32_16x16x4_f32(
              false, am, false, bm, (short)0, acc[mt], false, false);
        }
      }
    }

    if (layer == 0) {
      float badd = b1[n0 + ln];
      __syncthreads();                  // all waves done reading sH
      #pragma unroll
      for (int mt = 0; mt < 4; ++mt)
        #pragma unroll
        for (int i = 0; i < 8; ++i) {   // C/D: row = i + 8*hi, col = ln
          float v = acc[mt][i] + badd;
          sH[mt * 16 + hi * 8 + i][n0 + ln] = fmaxf(v, 0.0f);
        }
      __syncthreads();                  // h1 visible for layer 2
    } else {
      float badd = b2[n0 + ln];
      float bf   = bias[n0 + ln];
      #pragma unroll
      for (int mt = 0; mt < 4; ++mt)
        #pragma unroll
        for (int i = 0; i < 8; ++i) {
          float v = fmaxf(acc[mt][i] + badd, 0.0f);
          v = fmaxf(v + bf, 0.0f);
          int g = row0 + mt * 16 + hi * 8 + i;
          if (g < n) *(AS1 float*)(hio + (size_t)g * D + n0 + ln) = v;
        }
    }
  }
}

// ---------------------------------------------------------------------------
extern "C" void kernel_launch(void* const* d_in, const int* in_sizes, int n_in,
                              void* d_out, int out_size, void* d_ws, size_t ws_size,
                              hipStream_t stream) {
  const float* x    = (const float*)d_in[0];
  const int*   erow = (const int*)d_in[1];
  const int*   ecol = (const int*)d_in[2];
  const float* eval = (const float*)d_in[3];
  const float* W1   = (const float*)d_in[4];
  const float* b1   = (const float*)d_in[5];
  const float* W2   = (const float*)d_in[6];
  const float* b2   = (const float*)d_in[7];
  const float* eps  = (const float*)d_in[8];
  const float* bias = (const float*)d_in[9];
  float* out = (float*)d_out;

  int n  = in_sizes[0] / D;   // 100000 nodes
  int ne = in_sizes[1];       // 1.6M edges

  int total4 = n * (D / 4);
  gin_init<<<(total4 + 255) / 256, 256, 0, stream>>>(x, eps, out, total4);
  gin_scatter<<<(ne + 7) / 8, 256, 0, stream>>>(erow, ecol, eval, x, out, ne);
  gin_mlp<<<(n + ROWS - 1) / ROWS, 256, 0, stream>>>(out, W1, b1, W2, b2, bias, n);
}